// Projection_77816217469162
// MI455X (gfx1250) — compile-verified
//
#include <hip/hip_runtime.h>

typedef __attribute__((ext_vector_type(2))) float v2f;
typedef __attribute__((ext_vector_type(8))) float v8f;

#define IMG_W 3840
#define IMG_H 2160

// ---------------------------------------------------------------------------
// Kernel 1: zero-fill the output image (float4 streaming stores).
// ---------------------------------------------------------------------------
__global__ void proj_zero_kernel(float4* __restrict__ out4, int n4) {
    int idx = blockIdx.x * blockDim.x + threadIdx.x;
    int stride = gridDim.x * blockDim.x;
    float4 z = make_float4(0.f, 0.f, 0.f, 0.f);
    for (int i = idx; i < n4; i += stride) out4[i] = z;
}

// ---------------------------------------------------------------------------
// Kernel 2: fold K * inv(pose_next) * pose into a single 3x4 matrix M in ws.
// pose/pose_next are rigid [R t; 0 0 0 1], so
// inv(pose_next) = [R^T, -R^T t; 0 0 0 1].
// ---------------------------------------------------------------------------
__global__ void proj_prep_kernel(const float* __restrict__ pose,
                                 const float* __restrict__ pose_next,
                                 const float* __restrict__ Km,
                                 float* __restrict__ ws) {
    if (threadIdx.x != 0 || blockIdx.x != 0) return;

    float Pi[3][4];
    float t0 = pose_next[3], t1 = pose_next[7], t2 = pose_next[11];
    for (int i = 0; i < 3; ++i) {
        Pi[i][0] = pose_next[0 * 4 + i];
        Pi[i][1] = pose_next[1 * 4 + i];
        Pi[i][2] = pose_next[2 * 4 + i];
        Pi[i][3] = -(pose_next[0 * 4 + i] * t0 +
                     pose_next[1 * 4 + i] * t1 +
                     pose_next[2 * 4 + i] * t2);
    }

    float P2N[3][4];
    for (int i = 0; i < 3; ++i)
        for (int j = 0; j < 4; ++j)
            P2N[i][j] = Pi[i][0] * pose[0 * 4 + j] +
                        Pi[i][1] * pose[1 * 4 + j] +
                        Pi[i][2] * pose[2 * 4 + j] +
                        Pi[i][3] * pose[3 * 4 + j];

    for (int r = 0; r < 3; ++r)
        for (int c = 0; c < 4; ++c)
            ws[r * 4 + c] = Km[r * 3 + 0] * P2N[0][c] +
                            Km[r * 3 + 1] * P2N[1][c] +
                            Km[r * 3 + 2] * P2N[2][c];
}

// ---------------------------------------------------------------------------
// Kernel 3: per-wave, transform 16 pixels per iteration with
// V_WMMA_F32_16X16X4_F32 (A = M padded to 16x4, B = 16 homogeneous points),
// then scatter source depth to the projected integer pixel.
//
// A layout (32-bit 16x4, ISA 7.12.2): lane m (0-15) holds A[m][K=0],A[m][K=1];
//                                     lane m+16 holds A[m][K=2],A[m][K=3].
// B layout (4x16):                    lane n (0-15) holds B[K=0][n],B[K=1][n];
//                                     lane n+16 holds B[K=2][n],B[K=3][n].
// D layout (16x16 f32): VGPR r, lanes 0-15 -> D[r][n]. Rows 0..2 = (ud,vd,d).
// ---------------------------------------------------------------------------
__global__ void proj_wmma_kernel(const float* __restrict__ depth,
                                 const float* __restrict__ Km,
                                 const float* __restrict__ M,   // ws: 3x4
                                 float* __restrict__ out,
                                 int n_pix) {
    const int lane  = threadIdx.x & 31;
    const int wave  = (blockIdx.x * blockDim.x + threadIdx.x) >> 5;
    const int nwave = (gridDim.x * blockDim.x) >> 5;
    const int m     = lane & 15;
    const bool hi   = lane >= 16;

    const float fx = Km[0], cx = Km[2], fy = Km[4], cy = Km[5];
    const float inv_fx = 1.0f / fx, inv_fy = 1.0f / fy;

    // A matrix: rows 0..2 = M, rows 3..15 = 0. Constant across the loop.
    v2f a;
    a.x = (m < 3) ? M[m * 4 + (hi ? 2 : 0)] : 0.0f;
    a.y = (m < 3) ? M[m * 4 + (hi ? 3 : 1)] : 0.0f;

    const int n_grp = n_pix >> 4;   // groups of 16 pixels

    for (int g = wave; g < n_grp; g += nwave) {
        const int pi = (g << 4) + m;           // this lane's pixel (both halves)
        const int pc = pi < n_pix ? pi : n_pix - 1;
        const float d = depth[pc];

        // B column for point m: (X, Y) on lanes 0-15, (Z=d, 1) on 16-31.
        v2f b;
        if (!hi) {
            const int i = pc % IMG_W;
            const int j = pc / IMG_W;
            b.x = d * ((float)i - cx) * inv_fx;   // X
            b.y = d * ((float)j - cy) * inv_fy;   // Y
        } else {
            b.x = d;                              // Z
            b.y = 1.0f;                           // w
        }

        // uvd = M * [X Y Z 1]^T for 16 points. EXEC is all-1s here (loop
        // bound is wave-uniform, block size multiple of 32).
        v8f c = {};
        c = __builtin_amdgcn_wmma_f32_16x16x4_f32(
                /*neg_a=*/false, a, /*neg_b=*/false, b,
                /*c_mod=*/(short)0, c, /*reuse_a=*/false, /*reuse_b=*/false);

        // Lanes 0-15 hold D rows 0..2 for point n=lane: (u*d, v*d, d).
        if (!hi && pi < n_pix) {
            const float dd    = c[2];
            const float denom = fmaxf(dd, 0.0f) + 1e-12f;
            // 1-ULP v_rcp_f32 instead of the precise div_scale/div_fixup
            // chain: result only feeds a round-to-int on coords < 2^12, so
            // the relative error is far below the 0.5px rounding threshold.
            const float r  = __builtin_amdgcn_rcpf(denom);
            const float uf = c[0] * r;
            const float vf = c[1] * r;
            const int u = (int)rintf(uf);
            const int v = (int)rintf(vf);
            if (u >= 0 && u < IMG_W && v >= 0 && v < IMG_H) {
                out[v * IMG_W + u] = d;   // scatter, last-writer-wins
            }
        }
    }
}

// ---------------------------------------------------------------------------
// Launch: inputs are [depth(8294400), pose(16), pose_next(16), K(9)].
// ---------------------------------------------------------------------------
extern "C" void kernel_launch(void* const* d_in, const int* in_sizes, int n_in,
                              void* d_out, int out_size, void* d_ws, size_t ws_size,
                              hipStream_t stream) {
    const float* depth     = (const float*)d_in[0];
    const float* pose      = (const float*)d_in[1];
    const float* pose_next = (const float*)d_in[2];
    const float* Km        = (const float*)d_in[3];
    float* out = (float*)d_out;
    float* ws  = (float*)d_ws;

    const int n_pix = in_sizes[0];          // 2160*3840 = 8294400
    const int n4    = n_pix >> 2;

    proj_zero_kernel<<<2048, 256, 0, stream>>>((float4*)out, n4);
    proj_prep_kernel<<<1, 32, 0, stream>>>(pose, pose_next, Km, ws);
    proj_wmma_kernel<<<1024, 256, 0, stream>>>(depth, Km, ws, out, n_pix);
}